// RNNPredictor_19825569038602
// MI455X (gfx1250) — compile-verified
//
#include <hip/hip_runtime.h>
#include <hip/hip_bf16.h>
#include <math.h>

// ---------------------------------------------------------------------------
// Sizes (fixed by the reference)
// ---------------------------------------------------------------------------
#define BATCH 8192
#define TSTEP 24
#define NST   35
#define HID   512
#define MT    32            // batch rows per block
#define HSTR  516           // padded LDS stride for h (516 % 64 == 4 -> conflict free b64 loads)
#define XSTR  36            // padded x-tile stride (35 -> 36, col 35 stays zero)

typedef float v2f __attribute__((ext_vector_type(2)));
typedef float v8f __attribute__((ext_vector_type(8)));

static __device__ __forceinline__ v8f wmma_f32x4(v2f a, v2f b, v8f c) {
  // V_WMMA_F32_16X16X4_F32 : D[16x16] = A[16x4] * B[4x16] + C
  return __builtin_amdgcn_wmma_f32_16x16x4_f32(
      /*neg_a=*/false, a, /*neg_b=*/false, b,
      /*c_mod=*/(short)0, c, /*reuse_a=*/false, /*reuse_b=*/false);
}

static __device__ __forceinline__ float sigmoidf_(float v) {
  return 1.0f / (1.0f + __expf(-v));
}

// ---------------------------------------------------------------------------
// Kernel 1: per-block LSTM over all 24 timesteps for a 32-row batch tile.
//   gates[32,2048] = h[32,512] @ W_hh^T + x[32,35] @ W_ih^T + b_ih + b_hh
//   h,c update (VALU), then outputs[:,t,:] = h @ Wp^T + bp  (written to ws)
// All GEMMs via V_WMMA_F32_16X16X4_F32.
// ---------------------------------------------------------------------------
__global__ __launch_bounds__(256) void lstm_wmma_kernel(
    const float* __restrict__ x,    // [B, 24, 210]
    const float* __restrict__ Wih,  // [2048, 35]
    const float* __restrict__ Whh,  // [2048, 512]
    const float* __restrict__ bih,  // [2048]
    const float* __restrict__ bhh,  // [2048]
    const float* __restrict__ Wp,   // [35, 512]
    const float* __restrict__ bp,   // [35]
    float* __restrict__ outs)       // [B, 24, 35] (workspace)
{
  extern __shared__ float smem[];
  float* hbuf0 = smem;                              // [MT][HSTR]
  float* hbuf1 = smem + MT * HSTR;                  // [MT][HSTR]
  float* cbuf  = smem + 2 * MT * HSTR;              // [MT][HID]
  float* xt    = smem + 2 * MT * HSTR + MT * HID;   // [MT][XSTR]
  const int SMEM_FLOATS = 2 * MT * HSTR + MT * HID + MT * XSTR;

  const int tid   = threadIdx.x;
  const int lane  = tid & 31;
  const int wave  = tid >> 5;
  const int l16   = lane & 15;
  const int lhi   = lane >> 4;       // 0 or 1 (half-wave)
  const int bbase = blockIdx.x * MT;

  // zero h0, h1, c, xt (xt column 35 must stay zero forever)
  for (int i = tid; i < SMEM_FLOATS; i += 256) smem[i] = 0.0f;
  __syncthreads();

  for (int t = 0; t < TSTEP; ++t) {
    // ---- stage x tile (pm25 = feature 0 of 6) ----
    for (int i = tid; i < MT * NST; i += 256) {
      int m = i / NST, n = i - m * NST;
      xt[m * XSTR + n] = x[(size_t)((bbase + m) * TSTEP + t) * (NST * 6) + n * 6];
    }
    __syncthreads();

    float* hprev = (t & 1) ? hbuf1 : hbuf0;
    float* hcur  = (t & 1) ? hbuf0 : hbuf1;

    // ---- gate GEMM + LSTM cell ----
    for (int iu = 0; iu < 4; ++iu) {
      const int ntile = wave * 4 + iu;       // 0..31 -> 16 columns within [0,512)
      const int ncol  = ntile * 16 + l16;    // this lane's gate column

      v8f acc[4][2];                         // [gate i/f/g/o][mtile]
      #pragma unroll
      for (int j = 0; j < 4; ++j)
        #pragma unroll
        for (int mt = 0; mt < 2; ++mt)
          acc[j][mt] = (v8f){0.f,0.f,0.f,0.f,0.f,0.f,0.f,0.f};

      // A operand: h rows (LDS). lanes 0-15 -> K+{0,1}; lanes 16-31 -> K+{2,3}
      const float* hp0 = hprev + l16 * HSTR + 2 * lhi;
      const float* hp1 = hprev + (16 + l16) * HSTR + 2 * lhi;
      // B operand: W_hh rows (global, L2-resident)
      const float* wr0 = Whh + (size_t)(0 * 512 + ncol) * HID + 2 * lhi;
      const float* wr1 = Whh + (size_t)(1 * 512 + ncol) * HID + 2 * lhi;
      const float* wr2 = Whh + (size_t)(2 * 512 + ncol) * HID + 2 * lhi;
      const float* wr3 = Whh + (size_t)(3 * 512 + ncol) * HID + 2 * lhi;

      for (int k = 0; k < HID; k += 4) {
        v2f a0 = *(const v2f*)(hp0 + k);
        v2f a1 = *(const v2f*)(hp1 + k);
        v2f b0 = *(const v2f*)(wr0 + k);
        v2f b1 = *(const v2f*)(wr1 + k);
        v2f b2 = *(const v2f*)(wr2 + k);
        v2f b3 = *(const v2f*)(wr3 + k);
        acc[0][0] = wmma_f32x4(a0, b0, acc[0][0]);
        acc[0][1] = wmma_f32x4(a1, b0, acc[0][1]);
        acc[1][0] = wmma_f32x4(a0, b1, acc[1][0]);
        acc[1][1] = wmma_f32x4(a1, b1, acc[1][1]);
        acc[2][0] = wmma_f32x4(a0, b2, acc[2][0]);
        acc[2][1] = wmma_f32x4(a1, b2, acc[2][1]);
        acc[3][0] = wmma_f32x4(a0, b3, acc[3][0]);
        acc[3][1] = wmma_f32x4(a1, b3, acc[3][1]);
      }

      // x @ W_ih^T contribution: K = 35 padded to 36 (xt col 35 == 0)
      const float* xp0 = xt + l16 * XSTR + 2 * lhi;
      const float* xp1 = xt + (16 + l16) * XSTR + 2 * lhi;
      #pragma unroll
      for (int k = 0; k < 36; k += 4) {
        v2f a0 = *(const v2f*)(xp0 + k);
        v2f a1 = *(const v2f*)(xp1 + k);
        const int kk = k + 2 * lhi;          // 0..34
        #pragma unroll
        for (int j = 0; j < 4; ++j) {
          const float* wrow = Wih + (size_t)(j * 512 + ncol) * NST;
          v2f b;
          b.x = wrow[kk];
          b.y = (kk < 34) ? wrow[kk + 1] : 0.0f;   // guard W_ih[n][35]
          acc[j][0] = wmma_f32x4(a0, b, acc[j][0]);
          acc[j][1] = wmma_f32x4(a1, b, acc[j][1]);
        }
      }

      // bias (depends only on column -> same for all 8 rows this lane owns)
      float bias[4];
      #pragma unroll
      for (int j = 0; j < 4; ++j)
        bias[j] = bih[j * 512 + ncol] + bhh[j * 512 + ncol];

      // LSTM cell: i,f,g,o live at identical (lane, reg) slots across acc[j]
      #pragma unroll
      for (int mt = 0; mt < 2; ++mt) {
        #pragma unroll
        for (int r = 0; r < 8; ++r) {
          const int m = mt * 16 + r + 8 * lhi;   // D layout: M = r + 8*(lane/16)
          const float gi = acc[0][mt][r] + bias[0];
          const float gf = acc[1][mt][r] + bias[1];
          const float gg = acc[2][mt][r] + bias[2];
          const float go = acc[3][mt][r] + bias[3];
          const int ci = m * HID + ncol;
          float cv = cbuf[ci];
          cv = sigmoidf_(gf) * cv + sigmoidf_(gi) * tanhf(gg);
          cbuf[ci] = cv;
          hcur[m * HSTR + ncol] = sigmoidf_(go) * tanhf(cv);
        }
      }
    }
    __syncthreads();

    // ---- projection: outputs[:,t,:] = h @ Wp^T + bp  (N=35 padded to 48) ----
    if (wave < 6) {
      const int mt = wave & 1;           // m tile 0/1
      const int pt = wave >> 1;          // n tile 0..2
      const int pn = pt * 16 + l16;      // 0..47
      const bool valid = (pn < NST);

      v8f pacc = (v8f){0.f,0.f,0.f,0.f,0.f,0.f,0.f,0.f};
      const float* hp  = hcur + (mt * 16 + l16) * HSTR + 2 * lhi;
      const float* wpr = Wp + (size_t)(valid ? pn : 0) * HID + 2 * lhi;

      for (int k = 0; k < HID; k += 4) {
        v2f a = *(const v2f*)(hp + k);
        v2f b = *(const v2f*)(wpr + k);
        if (!valid) { b.x = 0.0f; b.y = 0.0f; }
        pacc = wmma_f32x4(a, b, pacc);
      }
      if (valid) {
        const float bb = bp[pn];
        #pragma unroll
        for (int r = 0; r < 8; ++r) {
          const int m = mt * 16 + r + 8 * lhi;
          outs[(size_t)((bbase + m) * TSTEP + t) * NST + pn] = pacc[r] + bb;
        }
      }
    }
  }
}

// ---------------------------------------------------------------------------
// Kernel 2: PDE diffusion (6 fixed iterations) + final blend.
//   div_f = ((v_x + v_y) * pm25) @ sw  -> fold into per-row scalar vs[i]
//   pm_new[j] = p[j] + sum_i p[i]*(vs[i]*A[i][j] + B[i][j]) + C[j]
//   A = -s1*dt*sw, B = -s2*k*dt*L, C = s3*dt*h_param  (precomputed in LDS)
// ---------------------------------------------------------------------------
__global__ __launch_bounds__(256) void pde_kernel(
    const float* __restrict__ x,        // [B, 24, 210]
    const float* __restrict__ adj,      // [35, 35]
    const float* __restrict__ lstm_out, // [B, 24, 35] (workspace)
    const float* __restrict__ s1p, const float* __restrict__ s2p,
    const float* __restrict__ s3p, const float* __restrict__ alp,
    const float* __restrict__ bep, const float* __restrict__ hpar,
    float* __restrict__ out)            // [B, 24, 35]
{
  __shared__ float Am[NST * NST];
  __shared__ float Bm[NST * NST];
  __shared__ float Cv[NST];

  const float dt = 3.0f, kc = 0.3f;
  const float s1 = s1p[0], s2 = s2p[0], s3 = s3p[0];
  const float al = alp[0], be = bep[0];

  for (int idx = threadIdx.x; idx < NST * NST; idx += 256) {
    const int i = idx / NST, j = idx - i * NST;
    float wij = adj[i * NST + j]; wij = (wij < 0.001f) ? 0.0f : wij;
    float wji = adj[j * NST + i]; wji = (wji < 0.001f) ? 0.0f : wji;
    const float sw = wji - wij;           // (w^T - w)[i][j]
    float Lij = -wij;
    if (i == j) {
      float rs = 0.0f;
      for (int q = 0; q < NST; ++q) {
        float wq = adj[i * NST + q];
        rs += (wq < 0.001f) ? 0.0f : wq;
      }
      Lij += rs;
    }
    Am[idx] = -s1 * dt * sw;
    Bm[idx] = -s2 * kc * dt * Lij;
  }
  for (int idx = threadIdx.x; idx < NST; idx += 256)
    Cv[idx] = s3 * dt * hpar[idx];
  __syncthreads();

  const int row = blockIdx.x * 256 + threadIdx.x;   // (b*24 + t)
  if (row >= BATCH * TSTEP) return;

  const float* xr   = x + (size_t)row * (NST * 6);
  const float* orow = lstm_out + (size_t)row * NST;

  float p[NST], vs[NST];
  for (int n = 0; n < NST; ++n) {
    p[n] = orow[n];
    const float sp = xr[n * 6 + 5];   // speed  (feature -1)
    const float dr = xr[n * 6 + 4];   // direction (feature -2)
    vs[n] = sp * (cosf(dr) + sinf(dr));
  }

  for (int it = 0; it < 6; ++it) {
    float np[NST];
    for (int j = 0; j < NST; ++j) {
      float acc = p[j] + Cv[j];
      for (int i2 = 0; i2 < NST; ++i2)
        acc += p[i2] * (vs[i2] * Am[i2 * NST + j] + Bm[i2 * NST + j]);  // LDS broadcast
      np[j] = acc;
    }
    for (int j = 0; j < NST; ++j) p[j] = np[j];
  }

  float* orow_out = out + (size_t)row * NST;
  for (int j = 0; j < NST; ++j) {
    const float pv = p[j] > 0.0f ? p[j] : 0.0f;
    orow_out[j] = al * orow[j] + be * pv;
  }
}

// ---------------------------------------------------------------------------
// Launch
// ---------------------------------------------------------------------------
extern "C" void kernel_launch(void* const* d_in, const int* in_sizes, int n_in,
                              void* d_out, int out_size, void* d_ws, size_t ws_size,
                              hipStream_t stream) {
  const float* x     = (const float*)d_in[0];
  const float* adj   = (const float*)d_in[1];
  const float* W_ih  = (const float*)d_in[2];
  const float* W_hh  = (const float*)d_in[3];
  const float* b_ih  = (const float*)d_in[4];
  const float* b_hh  = (const float*)d_in[5];
  const float* Wp    = (const float*)d_in[6];
  const float* bp    = (const float*)d_in[7];
  const float* s1    = (const float*)d_in[8];
  const float* s2    = (const float*)d_in[9];
  const float* s3    = (const float*)d_in[10];
  const float* alpha = (const float*)d_in[11];
  const float* beta  = (const float*)d_in[12];
  const float* hpar  = (const float*)d_in[13];
  float* out     = (float*)d_out;
  float* outs_ws = (float*)d_ws;   // [B, 24, 35] intermediate LSTM outputs

  const size_t shmem = (size_t)(2 * MT * HSTR + MT * HID + MT * XSTR) * sizeof(float);
  (void)hipFuncSetAttribute((const void*)lstm_wmma_kernel,
                            hipFuncAttributeMaxDynamicSharedMemorySize, (int)shmem);

  lstm_wmma_kernel<<<BATCH / MT, 256, shmem, stream>>>(
      x, W_ih, W_hh, b_ih, b_hh, Wp, bp, outs_ws);

  const int rows = BATCH * TSTEP;
  pde_kernel<<<(rows + 255) / 256, 256, 0, stream>>>(
      x, adj, outs_ws, s1, s2, s3, alpha, beta, hpar, out);
}